// SparseAutoencoder_57715770523660
// MI455X (gfx1250) — compile-verified
//
#include <hip/hip_runtime.h>
#include <hip/hip_bf16.h>
#include <float.h>
#include <math.h>
#include <stdint.h>

#define IN_DIM 30522
#define HID    8192
#define BATCH  2048
#define TOPK   32
#define SPARSITY_COEF 0.001f

typedef float v2f __attribute__((ext_vector_type(2)));
typedef float v8f __attribute__((ext_vector_type(8)));

// ---------------------------------------------------------------------------
// Encoder GEMM: hidden_pre = x @ W_enc + b_enc   (WMMA f32 16x16x4)
// Block: 256 threads = 8 waves; block tile 128(M) x 64(N); wave w owns rows
// [w*16, w*16+16). K staged in LDS chunks of 32 with double-buffered
// GLOBAL_LOAD_ASYNC_TO_LDS copies (ASYNCcnt pipeline). Both tiles are stored
// k-contiguous per row (A: [m][k], B transposed: [n][k]) so every WMMA
// fragment is a single 8-byte DS load (no register marshalling).
// ---------------------------------------------------------------------------
constexpr int KT         = 32;
constexpr int NCHUNK     = IN_DIM / KT;        // 953 full chunks
constexpr int KTAIL0     = NCHUNK * KT;        // 30496
constexpr int AS_STRIDE  = 36;                 // As[m][k] row stride (dwords)
constexpr int BT_STRIDE  = 36;                 // Bt[n][k] row stride (dwords)

__global__ __launch_bounds__(256) void enc_gemm_kernel(
    const float* __restrict__ x, const float* __restrict__ W,
    const float* __restrict__ bias, float* __restrict__ out)
{
  __shared__ float As2[2][128 * AS_STRIDE];   // 36,864 B
  __shared__ float Bt2[2][64 * BT_STRIDE];    // 18,432 B

  const int t    = threadIdx.x;
  const int wave = t >> 5;
  const int lane = t & 31;
  const int half = lane >> 4;
  const int lm   = lane & 15;

  // grid.x = 1024 = 8 (M sub-blocks of 128) x 128 (N blocks of 64); grid.y = 2
  const int m_sub = blockIdx.x >> 7;
  const int n_idx = blockIdx.x & 127;
  const int mBase = blockIdx.y * 1024 + m_sub * 128;
  const int nBase = n_idx * 64;
  const int wm    = wave * 16;

  v8f acc0 = {}; v8f acc1 = {}; v8f acc2 = {}; v8f acc3 = {};

  // ---- async stage of one K chunk into buffer bufi ----
  auto issue = [&](int bufi, int k0) {
    // x tile: 128 rows x 32 k, row-major, B64 granules (x row pitch is 8B aligned)
    #pragma unroll
    for (int e = 0; e < 8; ++e) {
      int lin = t + e * 256;            // 0..2047
      int row = lin >> 4;               // 0..127
      int kc  = lin & 15;               // k pair index
      unsigned goff = (unsigned)(((mBase + row) * IN_DIM + k0) * 4) + (unsigned)(kc * 8);
      unsigned lds  = (unsigned)(uintptr_t)&As2[bufi][row * AS_STRIDE + kc * 2];
      asm volatile("global_load_async_to_lds_b64 %0, %1, %2"
                   :: "v"(lds), "v"(goff), "s"(x) : "memory");
    }
    // W tile: 32 k-rows x 64 n, transposed into Bt[n][k]; per-lane B32 asyncs,
    // global side coalesced along n, LDS side scattered (async engine handles it)
    #pragma unroll
    for (int e = 0; e < 8; ++e) {
      int lin = t + e * 256;            // 0..2047
      int nn  = lin & 63;
      int kk  = lin >> 6;               // 0..31
      unsigned goff = (unsigned)(((k0 + kk) * HID + nBase + nn) * 4);
      unsigned lds  = (unsigned)(uintptr_t)&Bt2[bufi][nn * BT_STRIDE + kk];
      asm volatile("global_load_async_to_lds_b32 %0, %1, %2"
                   :: "v"(lds), "v"(goff), "s"(W) : "memory");
    }
  };

  // ---- WMMA over one staged chunk; every fragment is a contiguous b64 load ----
  auto compute = [&](int bi) {
    const float* Ar = &As2[bi][(wm + lm) * AS_STRIDE];
    const float* B0 = &Bt2[bi][(lm +  0) * BT_STRIDE];
    const float* B1 = &Bt2[bi][(lm + 16) * BT_STRIDE];
    const float* B2 = &Bt2[bi][(lm + 32) * BT_STRIDE];
    const float* B3 = &Bt2[bi][(lm + 48) * BT_STRIDE];
    #pragma unroll
    for (int kk = 0; kk < KT; kk += 4) {
      const int ka = kk + 2 * half;     // f32 A/B layout: VGPR0=K{0|2}, VGPR1=K{1|3}
      v2f a;  a[0]  = Ar[ka]; a[1]  = Ar[ka + 1];
      v2f b0; b0[0] = B0[ka]; b0[1] = B0[ka + 1];
      v2f b1; b1[0] = B1[ka]; b1[1] = B1[ka + 1];
      v2f b2; b2[0] = B2[ka]; b2[1] = B2[ka + 1];
      v2f b3; b3[0] = B3[ka]; b3[1] = B3[ka + 1];
      acc0 = __builtin_amdgcn_wmma_f32_16x16x4_f32(false, a, false, b0, (short)0, acc0, false, false);
      acc1 = __builtin_amdgcn_wmma_f32_16x16x4_f32(false, a, false, b1, (short)0, acc1, false, false);
      acc2 = __builtin_amdgcn_wmma_f32_16x16x4_f32(false, a, false, b2, (short)0, acc2, false, false);
      acc3 = __builtin_amdgcn_wmma_f32_16x16x4_f32(false, a, false, b3, (short)0, acc3, false, false);
    }
  };

  // ---- pipelined main loop over 953 full chunks ----
  issue(0, 0);
  for (int i = 0; i < NCHUNK; ++i) {
    const int bi = i & 1;
    if (i + 1 < NCHUNK) {
      issue(bi ^ 1, (i + 1) * KT);
      // 16 asyncs just issued for next buf; wait for the 16 of the current buf
      asm volatile("s_wait_asynccnt 0x10" ::: "memory");
    } else {
      asm volatile("s_wait_asynccnt 0x0" ::: "memory");
    }
    __syncthreads();          // all waves' copies for buf bi landed
    compute(bi);
    __syncthreads();          // all waves done reading buf bi before overwrite
  }

  // ---- synchronous zero-filled tail chunk (k = 30496 .. 30521, 26 valid) ----
  {
    const int tb = 0;
    for (int i = t; i < 128 * AS_STRIDE; i += 256) As2[tb][i] = 0.0f;
    for (int i = t; i < 64 * BT_STRIDE;  i += 256) Bt2[tb][i] = 0.0f;
    __syncthreads();
    #pragma unroll
    for (int e = 0; e < 16; ++e) {
      int lin = t + e * 256;
      int kk  = lin & 31;
      int mm  = lin >> 5;
      if (KTAIL0 + kk < IN_DIM)
        As2[tb][mm * AS_STRIDE + kk] = x[(size_t)(mBase + mm) * IN_DIM + KTAIL0 + kk];
    }
    #pragma unroll
    for (int e = 0; e < 8; ++e) {
      int lin = t + e * 256;
      int nn  = lin & 63;
      int kk  = lin >> 6;
      if (KTAIL0 + kk < IN_DIM)
        Bt2[tb][nn * BT_STRIDE + kk] = W[(size_t)(KTAIL0 + kk) * HID + nBase + nn];
    }
    __syncthreads();
    compute(tb);
  }

  // C layout: VGPR r holds row (r | r+8) per lane half, N = lane%16
  #pragma unroll
  for (int r = 0; r < 8; ++r) {
    const int row = mBase + wm + r + 8 * half;
    const int c0  = nBase + lm;
    out[(size_t)row * HID + c0     ] = acc0[r] + bias[c0     ];
    out[(size_t)row * HID + c0 + 16] = acc1[r] + bias[c0 + 16];
    out[(size_t)row * HID + c0 + 32] = acc2[r] + bias[c0 + 32];
    out[(size_t)row * HID + c0 + 48] = acc3[r] + bias[c0 + 48];
  }
}

// ---------------------------------------------------------------------------
// Per-hidden-feature decoder row norms: inv_norm[h] = 1 / max(||W_dec[h,:]||, 1e-12)
// ---------------------------------------------------------------------------
__global__ __launch_bounds__(256) void dec_norm_kernel(
    const float* __restrict__ Wd, float* __restrict__ inv_norm)
{
  __shared__ float red[256];
  const int h = blockIdx.x;
  const float* row = Wd + (size_t)h * IN_DIM;
  float s = 0.0f;
  for (int i = threadIdx.x; i < IN_DIM; i += 256) { float v = row[i]; s = fmaf(v, v, s); }
  red[threadIdx.x] = s;
  __syncthreads();
  for (int o = 128; o > 0; o >>= 1) {
    if (threadIdx.x < o) red[threadIdx.x] += red[threadIdx.x + o];
    __syncthreads();
  }
  if (threadIdx.x == 0) {
    float n = fmaxf(sqrtf(red[0]), 1e-12f);
    inv_norm[h] = 1.0f / n;
  }
}

// ---------------------------------------------------------------------------
// Top-K(32) per row: sparsify hidden in place, record (idx,val), accumulate
// sparsity |.| sum and active-feature count.
// ---------------------------------------------------------------------------
__global__ __launch_bounds__(256) void topk_kernel(
    float* __restrict__ hidden, int* __restrict__ t_idx,
    float* __restrict__ t_val, float* __restrict__ accs)
{
  __shared__ float sv[HID];
  __shared__ float rv[256];
  __shared__ int   ri[256];
  const int b = blockIdx.x;
  const int t = threadIdx.x;
  float* row = hidden + (size_t)b * HID;

  for (int i = t; i < HID; i += 256) sv[i] = row[i];
  __syncthreads();

  for (int j = 0; j < TOPK; ++j) {
    float bv = -FLT_MAX; int bi = HID;
    for (int i = t; i < HID; i += 256) {
      float v = sv[i];
      if (v > bv || (v == bv && i < bi)) { bv = v; bi = i; }
    }
    rv[t] = bv; ri[t] = bi;
    __syncthreads();
    for (int o = 128; o > 0; o >>= 1) {
      if (t < o) {
        float ov = rv[t + o]; int oi = ri[t + o];
        if (ov > rv[t] || (ov == rv[t] && oi < ri[t])) { rv[t] = ov; ri[t] = oi; }
      }
      __syncthreads();
    }
    if (t == 0) {
      t_idx[b * TOPK + j] = ri[0];
      t_val[b * TOPK + j] = rv[0];
      sv[ri[0]] = -FLT_MAX;
    }
    __syncthreads();
  }

  for (int i = t; i < HID; i += 256) row[i] = 0.0f;
  __syncthreads();
  if (t < TOPK) row[t_idx[b * TOPK + t]] = t_val[b * TOPK + t];
  if (t == 0) {
    float asum = 0.0f, cnt = 0.0f;
    for (int j = 0; j < TOPK; ++j) {
      float v = t_val[b * TOPK + j];
      asum += fabsf(v);
      if (fabsf(v) > 1e-6f) cnt += 1.0f;
    }
    atomicAdd(&accs[1], asum);
    atomicAdd(&accs[2], cnt);
  }
}

// ---------------------------------------------------------------------------
// Sparse decode: recon[b, tile*1024 .. +1024) = b_dec + sum_j val_j * Wn[idx_j, :]
// grid = (BATCH fastest, 30 col tiles) so each 32MB W_dec column slice lives in
// L2 across all 2048 batch blocks. Also accumulates sum((recon - x)^2).
// ---------------------------------------------------------------------------
__global__ __launch_bounds__(256) void decode_kernel(
    const float* __restrict__ x, const float* __restrict__ Wd,
    const float* __restrict__ b_dec, const int* __restrict__ t_idx,
    const float* __restrict__ t_val, const float* __restrict__ inv_nrm,
    float* __restrict__ recon, float* __restrict__ accs)
{
  __shared__ int   sidx[TOPK];
  __shared__ float sval[TOPK];
  __shared__ float red[256];
  const int b    = blockIdx.x;
  const int tile = blockIdx.y;
  const int t    = threadIdx.x;

  if (t < TOPK) {
    int id = t_idx[b * TOPK + t];
    sidx[t] = id;
    sval[t] = t_val[b * TOPK + t] * inv_nrm[id];
  }
  __syncthreads();

  float sq = 0.0f;
  #pragma unroll
  for (int e = 0; e < 4; ++e) {
    int col = tile * 1024 + e * 256 + t;
    if (col < IN_DIM) {
      float acc = b_dec[col];
      #pragma unroll 8
      for (int j = 0; j < TOPK; ++j)
        acc = fmaf(sval[j], Wd[(size_t)sidx[j] * IN_DIM + col], acc);
      recon[(size_t)b * IN_DIM + col] = acc;
      float d = acc - x[(size_t)b * IN_DIM + col];
      sq = fmaf(d, d, sq);
    }
  }
  red[t] = sq;
  __syncthreads();
  for (int o = 128; o > 0; o >>= 1) {
    if (t < o) red[t] += red[t + o];
    __syncthreads();
  }
  if (t == 0) atomicAdd(&accs[0], red[0]);
}

__global__ void init_kernel(float* __restrict__ accs)
{
  accs[0] = 0.0f; accs[1] = 0.0f; accs[2] = 0.0f;
}

__global__ void finalize_kernel(const float* __restrict__ accs, float* __restrict__ out4)
{
  float rec = accs[0] / ((float)BATCH * (float)IN_DIM);
  float sp  = SPARSITY_COEF * accs[1] / ((float)BATCH * (float)HID);
  out4[0] = rec + sp;               // loss
  out4[1] = rec;                    // reconstruction_loss
  out4[2] = sp;                     // sparsity_loss
  out4[3] = accs[2] / (float)BATCH; // active_features
}

// ---------------------------------------------------------------------------
extern "C" void kernel_launch(void* const* d_in, const int* in_sizes, int n_in,
                              void* d_out, int out_size, void* d_ws, size_t ws_size,
                              hipStream_t stream)
{
  const float* x     = (const float*)d_in[0];
  const float* W_enc = (const float*)d_in[1];
  const float* b_enc = (const float*)d_in[2];
  const float* W_dec = (const float*)d_in[3];
  const float* b_dec = (const float*)d_in[4];

  float* recon  = (float*)d_out;                   // [BATCH, IN_DIM]
  float* hidden = recon + (size_t)BATCH * IN_DIM;  // [BATCH, HID]
  float* scal   = hidden + (size_t)BATCH * HID;    // 4 scalar outputs

  char*  ws      = (char*)d_ws;
  int*   t_idx   = (int*)ws;                                        // BATCH*TOPK
  float* t_val   = (float*)(ws + (size_t)BATCH * TOPK * 4);         // BATCH*TOPK
  float* inv_nrm = (float*)(ws + (size_t)2 * BATCH * TOPK * 4);     // HID
  float* accs    = (float*)(ws + (size_t)2 * BATCH * TOPK * 4 + (size_t)HID * 4); // 3

  init_kernel<<<1, 1, 0, stream>>>(accs);
  enc_gemm_kernel<<<dim3(1024, 2), 256, 0, stream>>>(x, W_enc, b_enc, hidden);
  dec_norm_kernel<<<HID, 256, 0, stream>>>(W_dec, inv_nrm);
  topk_kernel<<<BATCH, 256, 0, stream>>>(hidden, t_idx, t_val, accs);
  decode_kernel<<<dim3(BATCH, 30), 256, 0, stream>>>(x, W_dec, b_dec, t_idx, t_val,
                                                     inv_nrm, recon, accs);
  finalize_kernel<<<1, 1, 0, stream>>>(accs, scal);
}